// BlazeEarEndToEndExportable_40398462386442
// MI455X (gfx1250) — compile-verified
//
#include <hip/hip_runtime.h>
#include <hip/hip_bf16.h>

#define N_ANCH   4000000
#define MAX_DET  1000
#define NBINS    4096
#define CAP      8192
#define SORT_N   8192
#define PADK     1024
#define IOU_THR  0.3f
#define CONF_THR 0.75f

typedef float v2f __attribute__((ext_vector_type(2)));
typedef float v8f __attribute__((ext_vector_type(8)));

// Order-preserving uint key of the clipped logit (monotone with sigmoid(clip(x))).
__device__ __forceinline__ unsigned okey(float s) {
    s = fminf(fmaxf(s, -100.0f), 100.0f);
    unsigned u = __float_as_uint(s);
    return (u & 0x80000000u) ? ~u : (u | 0x80000000u);
}

__global__ void k_init(unsigned* __restrict__ hist, unsigned* __restrict__ sel,
                       float* __restrict__ out) {
    int t = blockIdx.x * blockDim.x + threadIdx.x;
    int stride = blockDim.x * gridDim.x;
    for (int i = t; i < NBINS; i += stride) hist[i] = 0u;
    for (int i = t; i < 8; i += stride) sel[i] = 0u;
    for (int i = t; i < MAX_DET * 5; i += stride) out[i] = 0.0f;
}

// Pass 1: histogram of top 12 bits of the key. LDS-privatized, float4 stream.
__global__ void k_hist(const float4* __restrict__ sc4, int nv4,
                       unsigned* __restrict__ hist) {
    __shared__ unsigned lh[NBINS];
    for (int i = threadIdx.x; i < NBINS; i += blockDim.x) lh[i] = 0u;
    __syncthreads();
    int stride = gridDim.x * blockDim.x;
    for (int i = blockIdx.x * blockDim.x + threadIdx.x; i < nv4; i += stride) {
        __builtin_prefetch((const void*)(sc4 + i + stride), 0, 1); // global_prefetch_b8
        float4 v = sc4[i];
        atomicAdd(&lh[okey(v.x) >> 20], 1u);
        atomicAdd(&lh[okey(v.y) >> 20], 1u);
        atomicAdd(&lh[okey(v.z) >> 20], 1u);
        atomicAdd(&lh[okey(v.w) >> 20], 1u);
    }
    __syncthreads();
    for (int i = threadIdx.x; i < NBINS; i += blockDim.x)
        if (lh[i]) atomicAdd(&hist[i], lh[i]);
}

// Pass 2: find bin threshold whose suffix count first reaches MAX_DET.
__global__ void k_scan(const unsigned* __restrict__ hist, unsigned* __restrict__ sel) {
    if (threadIdx.x == 0 && blockIdx.x == 0) {
        unsigned acc = 0; int b = 0;
        for (int i = NBINS - 1; i >= 0; --i) {
            acc += hist[i];
            if (acc >= (unsigned)MAX_DET) { b = i; break; }
        }
        sel[0] = ((unsigned)b) << 20; // key >= this => candidate (superset of top-1000)
    }
}

// Pass 3: compact candidate (key, ~idx) pairs. Ties -> lowest index first.
__global__ void k_compact(const float4* __restrict__ sc4, int nv4,
                          const unsigned* __restrict__ sel, unsigned* __restrict__ cnt,
                          unsigned long long* __restrict__ cand) {
    unsigned thr = sel[0];
    int stride = gridDim.x * blockDim.x;
    for (int i = blockIdx.x * blockDim.x + threadIdx.x; i < nv4; i += stride) {
        float4 v = sc4[i];
        float vs[4] = {v.x, v.y, v.z, v.w};
#pragma unroll
        for (int c = 0; c < 4; ++c) {
            unsigned key = okey(vs[c]);
            if (key >= thr) {
                unsigned p = atomicAdd(cnt, 1u);
                if (p < (unsigned)CAP) {
                    unsigned idx = (unsigned)(i * 4 + c);
                    cand[p] = ((unsigned long long)key << 32) |
                              (unsigned long long)(~idx);
                }
            }
        }
    }
}

// Single-block bitonic sort (descending by key, index-ascending on ties),
// then gather+decode boxes for the top min(nc,1000).
__global__ void __launch_bounds__(1024)
k_sort(const unsigned long long* __restrict__ cand, unsigned* __restrict__ sel,
       const float* __restrict__ raw_scores, const float* __restrict__ raw_boxes,
       const float* __restrict__ anchors, float4* __restrict__ topBox,
       float* __restrict__ topSig) {
    __shared__ unsigned long long keys[SORT_N];
    int tid = threadIdx.x;
    int nc = (int)min(sel[1], (unsigned)CAP);
    for (int i = tid; i < SORT_N; i += 1024) keys[i] = (i < nc) ? cand[i] : 0ull;
    __syncthreads();
    for (int k = 2; k <= SORT_N; k <<= 1) {
        for (int j = k >> 1; j > 0; j >>= 1) {
            for (int i = tid; i < SORT_N; i += 1024) {
                int ixj = i ^ j;
                if (ixj > i) {
                    unsigned long long a = keys[i], b = keys[ixj];
                    bool up = ((i & k) == 0);
                    if (up ? (a < b) : (a > b)) { keys[i] = b; keys[ixj] = a; }
                }
            }
            __syncthreads();
        }
    }
    int ntop = min(nc, MAX_DET);
    if (tid < PADK) {
        float4 bx = make_float4(0.f, 0.f, 0.f, 0.f);
        float sg = -1.0f;
        if (tid < ntop) {
            unsigned long long kv = keys[tid];
            unsigned idx = ~(unsigned)(kv & 0xFFFFFFFFull);
            float s = fminf(fmaxf(raw_scores[idx], -100.0f), 100.0f);
            sg = 1.0f / (1.0f + expf(-s));
            const float* rb = raw_boxes + 4ull * idx;
            const float* an = anchors + 4ull * idx;
            float xc = rb[0] * 0.0078125f * an[2] + an[0];
            float yc = rb[1] * 0.0078125f * an[3] + an[1];
            float w  = rb[2] * 0.0078125f * an[2];
            float h  = rb[3] * 0.0078125f * an[3];
            float y0 = yc - h * 0.5f, y1v = yc + h * 0.5f;
            float x0 = xc - w * 0.5f, x1v = xc + w * 0.5f;
            bx.x = fminf(y0, y1v); bx.y = fminf(x0, x1v);
            bx.z = fmaxf(y0, y1v); bx.w = fmaxf(x0, x1v);
        }
        topBox[tid] = bx;
        topSig[tid] = sg;
        if (tid == 0) sel[2] = (unsigned)ntop;
    }
}

// Grid-parallel suppression-bitmask build. Block b owns rows [16b,16b+16).
// Bilinear denominator term areas[i]+areas[j] via v_wmma_f32_16x16x4_f32;
// predicates bit-packed into 32-bit column words; j<=i triangle masked here.
__global__ void __launch_bounds__(256)
k_supbits(const float4* __restrict__ topBox, unsigned* __restrict__ supW) {
    __shared__ float sy1[PADK], sx1[PADK], sy2[PADK], sx2[PADK], sar[PADK];
    __shared__ unsigned wbits[16 * 32];
    int tid = threadIdx.x;
    for (int r = tid; r < PADK; r += 256) {
        float4 b = topBox[r];
        sy1[r] = b.x; sx1[r] = b.y; sy2[r] = b.z; sx2[r] = b.w;
        sar[r] = (b.z - b.x) * (b.w - b.y);
    }
    for (int k = tid; k < 16 * 32; k += 256) wbits[k] = 0u;
    __syncthreads();

    int m0 = blockIdx.x * 16;
    unsigned lane = (unsigned)(tid & 31);
    unsigned wave = (unsigned)(tid >> 5);           // 8 waves
    bool lo = lane < 16u;
    int nn = (int)(lane & 15u);

    for (int t = (int)wave; t < PADK / 16; t += 8) { // uniform: 8 tiles per wave
        int n0 = t * 16;
        // A (16x4 f32): col0 = areas_I, col1 = ones, cols 2,3 = 0
        v2f a; a.x = lo ? sar[m0 + nn] : 0.0f; a.y = lo ? 1.0f : 0.0f;
        // B (4x16 f32): row0 = ones, row1 = areas_J, rows 2,3 = 0
        v2f b; b.x = lo ? 1.0f : 0.0f;         b.y = lo ? sar[n0 + nn] : 0.0f;
        v8f c = {};
        v8f d = __builtin_amdgcn_wmma_f32_16x16x4_f32(
            false, a, false, b, (short)0, c, false, false);
        int n = n0 + nn;
        int mbase = lo ? 0 : 8;
        float by1 = sy1[n], bxl = sx1[n], by2 = sy2[n], bxr = sx2[n];
        unsigned bit = 1u << (n & 31);
        int w = n >> 5;
#pragma unroll
        for (int r = 0; r < 8; ++r) {
            int m = m0 + mbase + r;
            float iy = fmaxf(fminf(sy2[m], by2) - fmaxf(sy1[m], by1), 0.0f);
            float ix = fmaxf(fminf(sx2[m], bxr) - fmaxf(sx1[m], bxl), 0.0f);
            float inter = iy * ix;
            float iou = inter / (d[r] - inter);     // NaN for padded rows -> no bit
            if (iou > IOU_THR) atomicOr(&wbits[(mbase + r) * 32 + w], bit);
        }
    }
    __syncthreads();
    // write out with j<=i triangle cleared (matches reference's idx>i condition)
    for (int k = tid; k < 16 * 32; k += 256) {
        int r = k >> 5, w = k & 31;
        int i = m0 + r;
        int wi = i >> 5;
        unsigned lowmask = 0xFFFFFFFFu >> (31 - (i & 31)); // bits 0..(i&31)
        unsigned word = wbits[k];
        word = (w > wi) ? word : ((w == wi) ? (word & ~lowmask) : 0u);
        supW[i * 32 + w] = word;
    }
}

// Stage 128KB bitmask into LDS (CDNA5: 320KB/WGP), then wave 0 runs the
// greedy recurrence in wave32 lockstep: keep bitmap in 1 VGPR/lane, rows
// applied with AND-NOT, group word re-broadcast via __shfl. Zero barriers
// in the serial chain. Then conf-filter, stable compaction, denorm, write.
__global__ void __launch_bounds__(1024)
k_final(const uint4* __restrict__ supW4, const float4* __restrict__ topBox,
        const float* __restrict__ topSig, const unsigned* __restrict__ sel,
        const float* __restrict__ scale_p, const float* __restrict__ pady_p,
        const float* __restrict__ padx_p, float* __restrict__ out) {
    __shared__ unsigned supLDS[PADK * 32];   // 128 KB
    __shared__ unsigned kW[32];
    __shared__ int posArr[PADK];
    int tid = threadIdx.x;

    uint4* s4 = (uint4*)supLDS;
    for (int k = tid; k < PADK * 32 / 4; k += 1024) s4[k] = supW4[k];
    __syncthreads();

    int ntop = (int)sel[2];
    if (tid < 32) {                                   // wave 0 only, lockstep
        int lane = tid;
        int base = lane * 32;
        unsigned keepw;
        if (ntop >= base + 32)      keepw = 0xFFFFFFFFu;
        else if (ntop <= base)      keepw = 0u;
        else                        keepw = 0xFFFFFFFFu >> (32 - (ntop - base));
        for (int g = 0; g < 32; ++g) {
            unsigned rowR[32];
#pragma unroll
            for (int r = 0; r < 32; ++r)
                rowR[r] = supLDS[(g * 32 + r) * 32 + lane];
            unsigned kg = (unsigned)__shfl((int)keepw, g, 32);
#pragma unroll
            for (int r = 0; r < 32; ++r) {
                if (kg & (1u << r)) {                 // uniform branch
                    keepw &= ~rowR[r];
                    kg = (unsigned)__shfl((int)keepw, g, 32);
                }
            }
        }
        kW[lane] = keepw;
    }
    __syncthreads();

    float sg = topSig[tid];
    int valid = (tid < ntop) && ((kW[tid >> 5] >> (tid & 31)) & 1u) &&
                (sg >= CONF_THR);
    posArr[tid] = valid;
    __syncthreads();
    if (tid == 0) {
        int p = 0;
        for (int r = 0; r < PADK; ++r) { int v = posArr[r]; posArr[r] = p; p += v; }
    }
    __syncthreads();
    if (valid) {
        float4 b = topBox[tid];
        float sc = *scale_p, py = *pady_p, px = *padx_p;
        int p = posArr[tid];
        out[p * 5 + 0] = b.x * sc * 256.0f - py;
        out[p * 5 + 1] = b.y * sc * 256.0f - px;
        out[p * 5 + 2] = b.z * sc * 256.0f - py;
        out[p * 5 + 3] = b.w * sc * 256.0f - px;
        out[p * 5 + 4] = sg;
    }
}

extern "C" void kernel_launch(void* const* d_in, const int* in_sizes, int n_in,
                              void* d_out, int out_size, void* d_ws, size_t ws_size,
                              hipStream_t stream) {
    (void)in_sizes; (void)n_in; (void)out_size; (void)ws_size;
    const float* raw_boxes  = (const float*)d_in[0];
    const float* raw_scores = (const float*)d_in[1];
    const float* anchors    = (const float*)d_in[2];
    const float* scale_p    = (const float*)d_in[3];
    const float* pady_p     = (const float*)d_in[4];
    const float* padx_p     = (const float*)d_in[5];
    float* out = (float*)d_out;

    char* ws = (char*)d_ws;
    unsigned* hist = (unsigned*)ws;                               // 16 KB
    unsigned* sel  = (unsigned*)(ws + 16384);                     // 32 B
    unsigned long long* cand = (unsigned long long*)(ws + 16416); // 64 KB
    float4* topBox = (float4*)(ws + 81952);                       // 16 KB
    float*  topSig = (float*)(ws + 98336);                        // 4 KB
    unsigned* supW = (unsigned*)(ws + 102432);                    // 128 KB

    int nv4 = N_ANCH / 4;
    k_init<<<8, 1024, 0, stream>>>(hist, sel, out);
    k_hist<<<1024, 256, 0, stream>>>((const float4*)raw_scores, nv4, hist);
    k_scan<<<1, 32, 0, stream>>>(hist, sel);
    k_compact<<<1024, 256, 0, stream>>>((const float4*)raw_scores, nv4, sel,
                                        sel + 1, cand);
    k_sort<<<1, 1024, 0, stream>>>(cand, sel, raw_scores, raw_boxes, anchors,
                                   topBox, topSig);
    k_supbits<<<64, 256, 0, stream>>>(topBox, supW);
    k_final<<<1, 1024, 0, stream>>>((const uint4*)supW, topBox, topSig, sel,
                                    scale_p, pady_p, padx_p, out);
}